// Binary_49950469653358
// MI455X (gfx1250) — compile-verified
//
#include <hip/hip_runtime.h>

#define D_WORLDS  32
#define N_WORLDS  512
#define BATCH     2048
#define K_DIM     64          // 2 * D_WORLDS
#define SW_STRIDE 68          // padded LDS row stride (floats): bank-conflict-free

typedef float v2f __attribute__((ext_vector_type(2)));
typedef float v8f __attribute__((ext_vector_type(8)));

// One workgroup (8 wave32) per batch element.
// Per wave: 64 world-columns, 2 M-tiles x 4 N-tiles of 16x16 f32 accumulators,
// K-loop of 16 steps of V_WMMA_F32_16X16X4_F32.
__global__ __launch_bounds__(256) void binop_l2norm_kernel(
    const float* __restrict__ states,   // [8*2048, 32, 512]
    const float* __restrict__ W,        // [128, 32, 64]
    const float* __restrict__ bias,     // [128, 32]
    const int*   __restrict__ indices,  // [2048]
    const int*   __restrict__ symbols,  // [2048]
    const int*   __restrict__ args,     // [2048, 2]
    float*       __restrict__ out)      // [2048, 32, 512]
{
    __shared__ float sW[D_WORLDS * SW_STRIDE];
    __shared__ float sb[D_WORLDS];

    const int n    = blockIdx.x;
    const int tid  = threadIdx.x;
    const int wave = tid >> 5;
    const int lane = tid & 31;
    const int half = lane >> 4;   // 0: lanes 0-15, 1: lanes 16-31
    const int c16  = lane & 15;

    const int sym = symbols[n];
    const int idx = indices[n];
    const int a0  = args[2 * n + 0];
    const int a1  = args[2 * n + 1];

    const float* lptr = states + (size_t)(a0 * BATCH + idx) * (D_WORLDS * N_WORLDS);
    const float* rptr = states + (size_t)(a1 * BATCH + idx) * (D_WORLDS * N_WORLDS);

    // ---- stage W[sym] (32x64 f32) into LDS with padded stride, and b[sym] ----
    const float* Wsym = W + (size_t)sym * (D_WORLDS * K_DIM);
    for (int i4 = tid; i4 < (D_WORLDS * K_DIM) / 4; i4 += 256) {
        float4 v = ((const float4*)Wsym)[i4];
        int e = i4 << 2;                       // flat element index
        *(float4*)&sW[(e >> 6) * SW_STRIDE + (e & 63)] = v;  // 16B-aligned (68%4==0)
    }
    if (tid < D_WORLDS) sb[tid] = bias[sym * D_WORLDS + tid];

    const int colbase = wave * 64;

    // prefetch the r slab's column block while we chew on l first
#pragma unroll
    for (int ci = 0; ci < 4; ++ci)
        __builtin_prefetch(rptr + (size_t)lane * N_WORLDS + colbase + ci * 16, 0, 0);

    __syncthreads();

    v8f acc[2][4];
#pragma unroll
    for (int mt = 0; mt < 2; ++mt)
#pragma unroll
        for (int nt = 0; nt < 4; ++nt)
            acc[mt][nt] = {};

    // ---- K loop: steps 0..7 read l (x rows 0..31), 8..15 read r (rows 32..63) ----
#pragma unroll
    for (int k = 0; k < 16; ++k) {
        const float* src = (k < 8) ? lptr : rptr;
        const int krow = ((k & 7) << 2) + (half << 1);   // row within the 32-row slab

        // A fragments: a[v] = Ws[mt*16 + c16][4k + 2*half + v]  (ds_load_b64)
        v2f afr[2];
#pragma unroll
        for (int mt = 0; mt < 2; ++mt)
            afr[mt] = *(const v2f*)&sW[(mt * 16 + c16) * SW_STRIDE + (k << 2) + (half << 1)];

#pragma unroll
        for (int nt = 0; nt < 4; ++nt) {
            // B fragment: b[v] = x[4k + 2*half + v][colbase + nt*16 + c16]
            const float* colp = src + (size_t)krow * N_WORLDS + colbase + nt * 16 + c16;
            v2f bfr;
            bfr.x = colp[0];
            bfr.y = colp[N_WORLDS];

            acc[0][nt] = __builtin_amdgcn_wmma_f32_16x16x4_f32(
                false, afr[0], false, bfr, (short)0, acc[0][nt], false, false);
            acc[1][nt] = __builtin_amdgcn_wmma_f32_16x16x4_f32(
                false, afr[1], false, bfr, (short)0, acc[1][nt], false, false);
        }
    }

    // ---- bias add: row M = mt*16 + 8*half + p ----
#pragma unroll
    for (int mt = 0; mt < 2; ++mt)
#pragma unroll
        for (int p = 0; p < 8; ++p) {
            float bv = sb[mt * 16 + (half << 3) + p];
#pragma unroll
            for (int nt = 0; nt < 4; ++nt)
                acc[mt][nt][p] += bv;
        }

    // ---- L2 normalize along d (32 rows / column) and store ----
    float* outp = out + (size_t)idx * (D_WORLDS * N_WORLDS);
#pragma unroll
    for (int nt = 0; nt < 4; ++nt) {
        float ss = 0.f;
#pragma unroll
        for (int mt = 0; mt < 2; ++mt)
#pragma unroll
            for (int p = 0; p < 8; ++p)
                ss += acc[mt][nt][p] * acc[mt][nt][p];
        // lane c holds rows {p+16mt}, lane c+16 holds rows {8+p+16mt}: combine halves
        ss += __shfl_xor(ss, 16, 32);
        float scale = rsqrtf(fmaxf(ss, 1e-12f));
#pragma unroll
        for (int mt = 0; mt < 2; ++mt)
#pragma unroll
            for (int p = 0; p < 8; ++p) {
                int row = mt * 16 + (half << 3) + p;
                outp[(size_t)row * N_WORLDS + colbase + nt * 16 + c16] =
                    acc[mt][nt][p] * scale;
            }
    }
}

extern "C" void kernel_launch(void* const* d_in, const int* in_sizes, int n_in,
                              void* d_out, int out_size, void* d_ws, size_t ws_size,
                              hipStream_t stream) {
    const float* states  = (const float*)d_in[0];
    const float* W       = (const float*)d_in[1];
    const float* bias    = (const float*)d_in[2];
    const int*   indices = (const int*)d_in[3];
    const int*   symbols = (const int*)d_in[4];
    const int*   args    = (const int*)d_in[5];
    float*       out     = (float*)d_out;

    binop_l2norm_kernel<<<BATCH, 256, 0, stream>>>(states, W, bias, indices,
                                                   symbols, args, out);
}